// net_RNN_41558103556869
// MI455X (gfx1250) — compile-verified
//
#include <hip/hip_runtime.h>

typedef __attribute__((ext_vector_type(2))) float v2f;
typedef __attribute__((ext_vector_type(8))) float v8f;

#define SEQ   2048
#define BATCH 4096
#define IN_F  4
#define HID_F 3

// ---------------------------------------------------------------------------
// Kernel 1: input projection via V_WMMA_F32_16X16X4_F32 (K=4 == IN_F).
//   rows r = t*BATCH + b  (x is [T,B,4] contiguous => row-major rows)
//   A[16x4]  = W_ih padded with zero rows (M>=3)
//   B[4x16]  = 16 consecutive x-rows as columns (N = lane%16)
//   D[h][n]  = pre[row n][h]  -> lanes 0..15 each store one 12B row.
//
// Software-pipelined: each wave owns 32 contiguous tiles, processed in groups
// of 4 with double-buffered loads so the next group's global_load_b64s are in
// flight while the current group's WMMAs execute (wait loadcnt<=4, not 0).
// pre is written into d_out (outs region) and consumed in place by kernel 2.
// ---------------------------------------------------------------------------
__global__ void rnn_preproj_wmma(const float* __restrict__ x,
                                 const float* __restrict__ W_ih,
                                 const float* __restrict__ b_ih,
                                 const float* __restrict__ b_hh,
                                 float* __restrict__ pre) {
  const int lane  = threadIdx.x & 31;
  const int wave  = (int)((blockIdx.x * blockDim.x + threadIdx.x) >> 5);
  const int m     = lane & 15;             // row (A) / column (B) index
  const int khalf = (lane >> 4) ? 2 : 0;   // lanes 16..31 hold K=2,3

  // A-matrix operand: lane m<3 holds W_ih[m][khalf], W_ih[m][khalf+1]
  v2f a; a.x = 0.0f; a.y = 0.0f;
  if (m < HID_F) {
    a.x = W_ih[m * IN_F + khalf];
    a.y = W_ih[m * IN_F + khalf + 1];
  }
  const float bias0 = b_ih[0] + b_hh[0];
  const float bias1 = b_ih[1] + b_hh[1];
  const float bias2 = b_ih[2] + b_hh[2];

  constexpr int TPW = 32;       // tiles per wave; launch must satisfy
                                // nWaves * TPW == SEQ*BATCH/16
  constexpr int G   = 4;        // tiles per pipeline group
  constexpr int NG  = TPW / G;  // 8 groups, fully unrolled

  const size_t row0 = (size_t)wave * TPW * 16 + m;
  const float* xb = x   + row0 * IN_F + khalf;   // + i*16*IN_F per tile
  float*       ob = pre + row0 * HID_F;          // + i*16*HID_F per tile

  v2f cur[G];
#pragma unroll
  for (int i = 0; i < G; ++i) {
    const float2 xv = *reinterpret_cast<const float2*>(xb + (size_t)i * 16 * IN_F);
    cur[i].x = xv.x; cur[i].y = xv.y;
  }

#pragma unroll
  for (int g = 0; g < NG; ++g) {
    v2f nxt[G];
    if (g + 1 < NG) {            // issue next group's loads before the WMMAs
#pragma unroll
      for (int i = 0; i < G; ++i) {
        const float2 xv = *reinterpret_cast<const float2*>(
            xb + ((size_t)(g + 1) * G + i) * 16 * IN_F);
        nxt[i].x = xv.x; nxt[i].y = xv.y;
      }
    }
    v8f d[G];
#pragma unroll
    for (int i = 0; i < G; ++i) {
      v8f c = {0.f, 0.f, 0.f, 0.f, 0.f, 0.f, 0.f, 0.f};
      d[i] = __builtin_amdgcn_wmma_f32_16x16x4_f32(
          /*neg_a=*/false, a, /*neg_b=*/false, cur[i],
          /*c_mod=*/(short)0, c, /*reuse_a=*/false, /*reuse_b=*/false);
    }
    if (lane < 16) {             // lanes 0..15 each store one 12B row per tile
#pragma unroll
      for (int i = 0; i < G; ++i) {
        float* o = ob + ((size_t)g * G + i) * 16 * HID_F;
        o[0] = d[i][0] + bias0;
        o[1] = d[i][1] + bias1;
        o[2] = d[i][2] + bias2;
      }
    }
    if (g + 1 < NG) {
#pragma unroll
      for (int i = 0; i < G; ++i) cur[i] = nxt[i];  // renamed away by unroll
    }
  }
}

// ---------------------------------------------------------------------------
// Kernel 2: serial scan over t, parallel over b. Reads pre from d_out (hot in
// the 192MB L2), overwrites in place with h_t, then writes h_last at the tail.
// 16 loads batched per unrolled block for memory-level parallelism.
// ---------------------------------------------------------------------------
__global__ void rnn_recurrence(const float* __restrict__ W_hh,
                               const float* __restrict__ h0,
                               float* __restrict__ out) {
  const int b = (int)(blockIdx.x * blockDim.x + threadIdx.x);  // 0..BATCH-1

  const float w00 = W_hh[0], w01 = W_hh[1], w02 = W_hh[2];
  const float w10 = W_hh[3], w11 = W_hh[4], w12 = W_hh[5];
  const float w20 = W_hh[6], w21 = W_hh[7], w22 = W_hh[8];

  float h0v = h0[(size_t)b * HID_F + 0];
  float h1v = h0[(size_t)b * HID_F + 1];
  float h2v = h0[(size_t)b * HID_F + 2];

  float* p = out + (size_t)b * HID_F;
  const size_t strideT = (size_t)BATCH * HID_F;

  constexpr int UN = 16;  // SEQ % UN == 0
  for (int t0 = 0; t0 < SEQ; t0 += UN) {
    float* pb = p + (size_t)t0 * strideT;
    float px[UN], py[UN], pz[UN];
#pragma unroll
    for (int j = 0; j < UN; ++j) {           // 16 independent loads in flight
      const float* q = pb + (size_t)j * strideT;
      px[j] = q[0]; py[j] = q[1]; pz[j] = q[2];
    }
    if (t0 + UN < SEQ)                       // -> global_prefetch_b8
      __builtin_prefetch(pb + (size_t)UN * strideT, 0, 3);
#pragma unroll
    for (int j = 0; j < UN; ++j) {           // serial 3x3 recurrent update
      const float n0 = fmaxf(px[j] + w00 * h0v + w01 * h1v + w02 * h2v, 0.0f);
      const float n1 = fmaxf(py[j] + w10 * h0v + w11 * h1v + w12 * h2v, 0.0f);
      const float n2 = fmaxf(pz[j] + w20 * h0v + w21 * h1v + w22 * h2v, 0.0f);
      h0v = n0; h1v = n1; h2v = n2;
      float* q = pb + (size_t)j * strideT;
      q[0] = n0; q[1] = n1; q[2] = n2;       // outs[t][b][:] = h_t (in place)
    }
  }
  float* hl = out + (size_t)SEQ * strideT + (size_t)b * HID_F;
  hl[0] = h0v; hl[1] = h1v; hl[2] = h2v;     // h_last
}

// ---------------------------------------------------------------------------
extern "C" void kernel_launch(void* const* d_in, const int* in_sizes, int n_in,
                              void* d_out, int out_size, void* d_ws, size_t ws_size,
                              hipStream_t stream) {
  (void)in_sizes; (void)n_in; (void)out_size; (void)d_ws; (void)ws_size;
  const float* x    = (const float*)d_in[0];  // [SEQ, BATCH, 4]
  const float* h0   = (const float*)d_in[1];  // [1, BATCH, 3]
  const float* W_ih = (const float*)d_in[2];  // [3, 4]
  const float* W_hh = (const float*)d_in[3];  // [3, 3]
  const float* b_ih = (const float*)d_in[4];  // [3]
  const float* b_hh = (const float*)d_in[5];  // [3]
  float* out = (float*)d_out;                 // [SEQ*BATCH*3] outs ++ [BATCH*3] h_last

  // Pass 1: bandwidth-bound WMMA projection, pre -> d_out (outs region).
  // 2048 blocks * 256 threads = 16384 waves * 32 tiles = SEQ*BATCH/16 tiles.
  rnn_preproj_wmma<<<dim3(2048), dim3(256), 0, stream>>>(x, W_ih, b_ih, b_hh, out);

  // Pass 2: latency-bound scan, 1 wave per block spread across WGPs.
  rnn_recurrence<<<dim3(BATCH / 32), dim3(32), 0, stream>>>(W_hh, h0, out);
}